// CustomGraphConv_43018392436835
// MI455X (gfx1250) — compile-verified
//
#include <hip/hip_runtime.h>

// ---------------------------------------------------------------------------
// CustomGraphConv for MI455X (gfx1250), compile-only tuned.
//
// msg_e = W_e @ x[src_e]   (per-edge 16x16 matvec, f32)
// out   = relu(scatter_add(msg, dst) + bias)
//
// HBM-bound: 1.64 GB of per-edge weights streamed once -> ~75us floor at
// 23.3 TB/s; compute (0.82 GFLOP) is free. Math via V_WMMA_F32_16X16X4_F32:
// one wave handles 16 edges, edge j occupies output column j. Per chain
// (edge j, K-chunk c):
//   A = W_j[:, 4c:4c+4]   (16x4 f32, one non-temporal b64 per lane)
//   B = x_j[4c:4c+4] placed in column j only (v_cndmask, zeros elsewhere)
// 64 WMMAs (4 independent accumulators, chains of 16) build the 16x16 D whose
// column n is msg of edge n. C/D layout: lane (n,h) holds D[v+8h, n] in VGPR v.
//
// Wave id is forced scalar via readfirstlane so the full-group path is
// straight-line (no EXEC juggling, no accumulator PHI copies).
// ---------------------------------------------------------------------------

typedef float v2f __attribute__((ext_vector_type(2)));
typedef float v8f __attribute__((ext_vector_type(8)));

#ifndef EDGE_BLOCK_THREADS
#define EDGE_BLOCK_THREADS 256
#endif

__device__ __forceinline__ v8f wmma_acc(v2f a, v2f b, v8f c) {
    return __builtin_amdgcn_wmma_f32_16x16x4_f32(
        /*neg_a=*/false, a, /*neg_b=*/false, b,
        /*c_mod=*/(short)0, c, /*reuse_a=*/false, /*reuse_b=*/false);
}

// --------------------------- kernel 1: zero accum ---------------------------
__global__ void cgc_zero_kernel(float* __restrict__ out, int n) {
    int i = blockIdx.x * blockDim.x + threadIdx.x;
    if (i < n) out[i] = 0.0f;
}

// ----------------------- kernel 2: per-edge WMMA pass -----------------------
__global__ __launch_bounds__(EDGE_BLOCK_THREADS)
void cgc_edge_wmma_kernel(const float* __restrict__ x,
                          const int*   __restrict__ edge_index, // [2, E]
                          const float* __restrict__ W,          // [E, 16, 16]
                          float*       __restrict__ accum,      // [N, 16]
                          int nEdges) {
    const int lane = threadIdx.x & 31;      // wave32
    const int n    = lane & 15;             // output column / edge slot
    const int h    = lane >> 4;             // K-half selector

    // Force the wave id into an SGPR so groupBase/rem are scalar.
    const int wave = __builtin_amdgcn_readfirstlane(
        blockIdx.x * (EDGE_BLOCK_THREADS / 32) + (threadIdx.x >> 5));
    const int groupBase = wave * 16;
    if (groupBase >= nEdges) return;        // scalar branch
    int rem = nEdges - groupBase;
    if (rem > 16) rem = 16;

    const float* Wg = W + (size_t)groupBase * 256;

    if (rem == 16) {
        // ------------------ fast path: straight line, no branches ----------
        const int src = edge_index[groupBase + n];
        const int dst = edge_index[(size_t)nEdges + groupBase + n];

        // x chunks for column n: xp[c] = { x[src*16+4c+2h], +1 }
        const float* xb = x + (size_t)src * 16 + 2 * h;
        v2f xp[4];
#pragma unroll
        for (int c = 0; c < 4; ++c)
            xp[c] = *(const v2f*)(xb + 4 * c);

        // Prefetch next group's 16 KB weight block (streaming hint).
        if (groupBase + 32 <= nEdges) {
#pragma unroll
            for (int k = 0; k < 4; ++k)
                __builtin_prefetch((const char*)(Wg + 16 * 256) + lane * 128 + k * 4096, 0, 1);
        }

        v8f acc[4];
#pragma unroll
        for (int q = 0; q < 4; ++q) acc[q] = (v8f){0, 0, 0, 0, 0, 0, 0, 0};

#pragma unroll
        for (int j = 0; j < 16; ++j) {
            // A source: lane (m=n, h) needs W_j[n, 4c + 2h .. +1]
            const float* Wj = Wg + (size_t)j * 256 + n * 16 + 2 * h;
#pragma unroll
            for (int c = 0; c < 4; ++c) {
                v2f a = __builtin_nontemporal_load((const v2f*)(Wj + 4 * c));
                v2f b;
                b.x = (n == j) ? xp[c].x : 0.0f;   // one v_cmp per j,
                b.y = (n == j) ? xp[c].y : 0.0f;   // dual cndmask per chunk
                acc[j & 3] = wmma_acc(a, b, acc[j & 3]);
            }
        }

        v8f d = (acc[0] + acc[1]) + (acc[2] + acc[3]);

        // Scatter-add: lane (n,h) owns rows v+8h of edge n's message.
        float* outp = accum + (size_t)dst * 16 + 8 * h;
#pragma unroll
        for (int v = 0; v < 8; ++v)
            atomicAdd(outp + v, d[v]);
    } else {
        // ------------- tail path (at most one wave): plain VALU ------------
        if (n < rem) {
            const int src = edge_index[groupBase + n];
            const int dst = edge_index[(size_t)nEdges + groupBase + n];
            const float* xv = x + (size_t)src * 16;
            const float* Wj = Wg + (size_t)n * 256 + 8 * h * 16;
            float* outp = accum + (size_t)dst * 16 + 8 * h;
#pragma unroll
            for (int r = 0; r < 8; ++r) {
                float s = 0.0f;
#pragma unroll
                for (int i = 0; i < 16; ++i)
                    s = fmaf(Wj[r * 16 + i], xv[i], s);
                atomicAdd(outp + r, s);
            }
        }
    }
}

// ----------------------- kernel 3: bias + ReLU (in place) -------------------
__global__ void cgc_bias_relu_kernel(float* __restrict__ out,
                                     const float* __restrict__ bias, int n) {
    int i = blockIdx.x * blockDim.x + threadIdx.x;
    if (i < n) out[i] = fmaxf(out[i] + bias[i & 15], 0.0f);
}

// ---------------------------------------------------------------------------
extern "C" void kernel_launch(void* const* d_in, const int* in_sizes, int n_in,
                              void* d_out, int out_size, void* d_ws, size_t ws_size,
                              hipStream_t stream) {
    (void)n_in; (void)d_ws; (void)ws_size;
    const float* x    = (const float*)d_in[0];
    const int*   ei   = (const int*)d_in[1];   // [2, E]
    // d_in[2] = edge_attr (unused by message(), matching reference)
    const float* W    = (const float*)d_in[3]; // [E, 16, 16]
    const float* bias = (const float*)d_in[4]; // [16]

    const int nEdges = in_sizes[2];            // edge_attr count == E
    const int nOut   = out_size;               // N_NODES * 16

    // 1) zero the accumulator (d_out doubles as accumulator; harness poisons it)
    cgc_zero_kernel<<<(nOut + 255) / 256, 256, 0, stream>>>((float*)d_out, nOut);

    // 2) edge pass: one wave per 16 edges
    const int waves   = (nEdges + 15) / 16;
    const int threads = waves * 32;
    const int blocks  = (threads + EDGE_BLOCK_THREADS - 1) / EDGE_BLOCK_THREADS;
    cgc_edge_wmma_kernel<<<blocks, EDGE_BLOCK_THREADS, 0, stream>>>(
        x, ei, W, (float*)d_out, nEdges);

    // 3) bias + ReLU in place
    cgc_bias_relu_kernel<<<(nOut + 255) / 256, 256, 0, stream>>>(
        (float*)d_out, bias, nOut);
}